// GraphEncoder_51771535786610
// MI455X (gfx1250) — compile-verified
//
#include <hip/hip_runtime.h>
#include <hip/hip_bf16.h>

typedef __attribute__((ext_vector_type(2))) float v2f;
typedef __attribute__((ext_vector_type(8))) float v8f;

#define NN 100000
#define EE 1600000
static constexpr float TAU_V = 0.25f;
static constexpr float EPS_V = 1e-12f;

// ---------------------------------------------------------------------------
// Dense GEMM via V_WMMA_F32_16X16X4_F32. One wave per 16x16 output tile.
// D = ACT( (A (+a_bias per col if ABIAS)) @ B (+d_bias per col if DBIAS) )
// ACT: 0=none, 1=leaky(0.01), 2=relu.  All flags compile-time -> branch-free
// K-loop: {3-load clause + v_wmma} per k-step, two accumulators to halve the
// WMMA RAW chain.
// ---------------------------------------------------------------------------
template <int K, int F, bool ABIAS, bool DBIAS, int ACT>
__global__ __launch_bounds__(256)
void gemm16(const float* __restrict__ A,
            const float* __restrict__ B,
            const float* __restrict__ a_bias,
            const float* __restrict__ d_bias,
            float* __restrict__ D,
            int nrows)
{
    const int tiles_n = F >> 4;
    const int wave    = (blockIdx.x * blockDim.x + threadIdx.x) >> 5;
    const int total   = (nrows >> 4) * tiles_n;
    if (wave >= total) return;                 // wave-uniform: EXEC stays all-ones
    const int tn   = wave % tiles_n;
    const int tm   = wave / tiles_n;
    const int lane = threadIdx.x & 31;
    const int hl   = lane >> 4;                // which half-wave
    const int l15  = lane & 15;
    const int m0   = tm << 4;
    const int n0   = tn << 4;

    const float* __restrict__ arow = A + (size_t)(m0 + l15) * K;

    v8f c0 = {}, c1 = {};
#pragma unroll
    for (int k = 0; k < K; k += 4) {
        const int k0 = k + (hl << 1);
        v2f a, b;
        a.x = arow[k0];
        a.y = arow[k0 + 1];
        if (ABIAS) { a.x += a_bias[k0]; a.y += a_bias[k0 + 1]; }
        b.x = B[(size_t)k0 * F + n0 + l15];
        b.y = B[(size_t)(k0 + 1) * F + n0 + l15];
        if ((k >> 2) & 1)
            c1 = __builtin_amdgcn_wmma_f32_16x16x4_f32(
                     false, a, false, b, (short)0, c1, false, false);
        else
            c0 = __builtin_amdgcn_wmma_f32_16x16x4_f32(
                     false, a, false, b, (short)0, c0, false, false);
    }
#pragma unroll
    for (int j = 0; j < 8; ++j) {
        const int row = m0 + j + (hl << 3);
        const int col = n0 + l15;
        float v = c0[j] + c1[j];
        if (DBIAS) v += d_bias[col];
        if (ACT == 1)      v = (v > 0.f) ? v : 0.01f * v;
        else if (ACT == 2) v = fmaxf(v, 0.f);
        D[(size_t)row * F + col] = v;
    }
}

// ---------------------------------------------------------------------------
// Row-wise L2 normalize, 64 floats per row. One wave per row (wave32).
// ---------------------------------------------------------------------------
__global__ __launch_bounds__(256)
void l2norm64(const float* __restrict__ src, float* __restrict__ dst, int n)
{
    const int wave = (blockIdx.x * blockDim.x + threadIdx.x) >> 5;
    if (wave >= n) return;
    const int lane = threadIdx.x & 31;
    const float2* p = (const float2*)(src + (size_t)wave * 64);
    float2 v = p[lane];
    float s = v.x * v.x + v.y * v.y;
#pragma unroll
    for (int off = 16; off; off >>= 1) s += __shfl_xor(s, off, 32);
    const float inv = 1.0f / fmaxf(sqrtf(s), EPS_V);
    float2 o; o.x = v.x * inv; o.y = v.y * inv;
    ((float2*)(dst + (size_t)wave * 64))[lane] = o;
}

// ---------------------------------------------------------------------------
// Per-edge: v = exp(dot(nr[dst], nl[src]))/TAU ; atomicMax per-src (v > 0).
// 8 lanes per edge.
// ---------------------------------------------------------------------------
__global__ __launch_bounds__(256)
void edge_cos(const float* __restrict__ nl,
              const float* __restrict__ nr,
              const int* __restrict__ src,
              const int* __restrict__ dst,
              float* __restrict__ vbuf,
              float* __restrict__ mbuf, int ne)
{
    const int gt = blockIdx.x * blockDim.x + threadIdx.x;
    const int e  = gt >> 3;
    if (e >= ne) return;
    const int sub = gt & 7;
    const int s = src[e], d = dst[e];
    const float4* pl = (const float4*)(nl + (size_t)s * 64);
    const float4* pr = (const float4*)(nr + (size_t)d * 64);
    const float4 a0 = pl[sub * 2], a1 = pl[sub * 2 + 1];
    const float4 b0 = pr[sub * 2], b1 = pr[sub * 2 + 1];
    float t = a0.x * b0.x + a0.y * b0.y + a0.z * b0.z + a0.w * b0.w
            + a1.x * b1.x + a1.y * b1.y + a1.z * b1.z + a1.w * b1.w;
    t += __shfl_xor(t, 4, 32);
    t += __shfl_xor(t, 2, 32);
    t += __shfl_xor(t, 1, 32);
    if (sub == 0) {
        const float vv = __expf(t) / TAU_V;   // strictly positive
        vbuf[e] = vv;
        atomicMax((unsigned int*)(mbuf + s), __float_as_uint(vv));
    }
}

// ---------------------------------------------------------------------------
// e = exp(v - m[src]);  den[src] += e  (f32 hardware atomic in L2)
// ---------------------------------------------------------------------------
__global__ __launch_bounds__(256)
void edge_exp_den(float* __restrict__ vbuf,
                  const float* __restrict__ mbuf,
                  const int* __restrict__ src,
                  float* __restrict__ den, int ne)
{
    const int e = blockIdx.x * blockDim.x + threadIdx.x;
    if (e >= ne) return;
    const int s = src[e];
    const float ee = __expf(vbuf[e] - mbuf[s]);
    vbuf[e] = ee;
    unsafeAtomicAdd(den + s, ee);
}

// ---------------------------------------------------------------------------
// a = e/den[src];  agg[dst] += x_l[src] * a.  8 lanes per edge, 8 f32 each.
// ---------------------------------------------------------------------------
__global__ __launch_bounds__(256)
void edge_scatter(const float* __restrict__ xl,
                  const float* __restrict__ ebuf,
                  const float* __restrict__ den,
                  const int* __restrict__ src,
                  const int* __restrict__ dst,
                  float* __restrict__ agg, int ne)
{
    const int gt = blockIdx.x * blockDim.x + threadIdx.x;
    const int e  = gt >> 3;
    if (e >= ne) return;
    const int sub = gt & 7;
    const int s = src[e], d = dst[e];
    const float a = ebuf[e] / den[s];
    const float4* px = (const float4*)(xl + (size_t)s * 64);
    const float4 x0 = px[sub * 2], x1 = px[sub * 2 + 1];
    float* po = agg + (size_t)d * 64 + sub * 8;
    unsafeAtomicAdd(po + 0, x0.x * a);
    unsafeAtomicAdd(po + 1, x0.y * a);
    unsafeAtomicAdd(po + 2, x0.z * a);
    unsafeAtomicAdd(po + 3, x0.w * a);
    unsafeAtomicAdd(po + 4, x1.x * a);
    unsafeAtomicAdd(po + 5, x1.y * a);
    unsafeAtomicAdd(po + 6, x1.z * a);
    unsafeAtomicAdd(po + 7, x1.w * a);
}

// ---------------------------------------------------------------------------
extern "C" void kernel_launch(void* const* d_in, const int* in_sizes, int n_in,
                              void* d_out, int out_size, void* d_ws, size_t ws_size,
                              hipStream_t stream)
{
    const float* x     = (const float*)d_in[0];
    const int*   ei    = (const int*)  d_in[1];
    const float* W_in  = (const float*)d_in[2];
    const float* b_in  = (const float*)d_in[3];
    const float* Wl1   = (const float*)d_in[4];
    const float* Wr1   = (const float*)d_in[5];
    const float* bias1 = (const float*)d_in[6];
    const float* Wc1   = (const float*)d_in[7];
    const float* Wl2   = (const float*)d_in[8];
    const float* Wr2   = (const float*)d_in[9];
    const float* bias2 = (const float*)d_in[10];
    const float* Wc2   = (const float*)d_in[11];
    const float* W3    = (const float*)d_in[12];
    const float* b3    = (const float*)d_in[13];
    const float* W4    = (const float*)d_in[14];
    const float* b4    = (const float*)d_in[15];
    const int* src = ei;
    const int* dst = ei + EE;

    float* ws = (float*)d_ws;
    size_t off = 0;
    float* h    = ws + off; off += (size_t)NN * 128;
    float* xl   = ws + off; off += (size_t)NN * 64;
    float* nl   = ws + off; off += (size_t)NN * 64;
    float* nr   = ws + off; off += (size_t)NN * 64;
    float* hA   = ws + off; off += (size_t)NN * 64;
    float* vbuf = ws + off; off += (size_t)EE;
    float* agg  = ws + off; off += (size_t)NN * 64;   // [agg | m | den] contiguous
    float* mbuf = ws + off; off += (size_t)NN;
    float* den  = ws + off; off += (size_t)NN;
    const size_t zero_bytes = (size_t)NN * 66 * sizeof(float);

    const dim3 B256(256);
    const int gemm_blk_128 = ((NN / 16) * 8 + 7) / 8;   // F=128 tiles / 8 waves
    const int gemm_blk_64  = ((NN / 16) * 4 + 7) / 8;   // F=64
    const int gemm_blk_32  = ((NN / 16) * 2 + 7) / 8;   // F=32
    const int norm_blk     = (NN * 32 + 255) / 256;
    const int e8_blk       = (EE * 8 + 255) / 256;
    const int e1_blk       = (EE + 255) / 256;

    // h = leaky(x @ W_in + b_in)
    gemm16<128, 128, false, true, 1><<<gemm_blk_128, B256, 0, stream>>>(
        x, W_in, nullptr, b_in, h, NN);

    // ---------------- GAT layer 1 (128 -> 64) ----------------
    gemm16<128, 64, false, false, 0><<<gemm_blk_64, B256, 0, stream>>>(
        h, Wl1, nullptr, nullptr, xl, NN);
    gemm16<128, 64, false, false, 0><<<gemm_blk_64, B256, 0, stream>>>(
        h, Wr1, nullptr, nullptr, nr, NN);
    l2norm64<<<norm_blk, B256, 0, stream>>>(xl, nl, NN);
    l2norm64<<<norm_blk, B256, 0, stream>>>(nr, nr, NN);
    hipMemsetAsync(agg, 0, zero_bytes, stream);
    edge_cos<<<e8_blk, B256, 0, stream>>>(nl, nr, src, dst, vbuf, mbuf, EE);
    edge_exp_den<<<e1_blk, B256, 0, stream>>>(vbuf, mbuf, src, den, EE);
    edge_scatter<<<e8_blk, B256, 0, stream>>>(xl, vbuf, den, src, dst, agg, EE);
    // h1 = leaky((agg + bias1) @ Wc1)
    gemm16<64, 64, true, false, 1><<<gemm_blk_64, B256, 0, stream>>>(
        agg, Wc1, bias1, nullptr, hA, NN);

    // ---------------- GAT layer 2 (64 -> 64) ----------------
    gemm16<64, 64, false, false, 0><<<gemm_blk_64, B256, 0, stream>>>(
        hA, Wl2, nullptr, nullptr, xl, NN);
    gemm16<64, 64, false, false, 0><<<gemm_blk_64, B256, 0, stream>>>(
        hA, Wr2, nullptr, nullptr, nr, NN);
    l2norm64<<<norm_blk, B256, 0, stream>>>(xl, nl, NN);
    l2norm64<<<norm_blk, B256, 0, stream>>>(nr, nr, NN);
    hipMemsetAsync(agg, 0, zero_bytes, stream);
    edge_cos<<<e8_blk, B256, 0, stream>>>(nl, nr, src, dst, vbuf, mbuf, EE);
    edge_exp_den<<<e1_blk, B256, 0, stream>>>(vbuf, mbuf, src, den, EE);
    edge_scatter<<<e8_blk, B256, 0, stream>>>(xl, vbuf, den, src, dst, agg, EE);
    // h2 = leaky((agg + bias2) @ Wc2)  -> reuse h buffer
    gemm16<64, 64, true, false, 1><<<gemm_blk_64, B256, 0, stream>>>(
        agg, Wc2, bias2, nullptr, h, NN);

    // ---------------- Head ----------------
    // t = relu(h2 @ W3 + b3) -> xl ; z = t @ W4 + b4 -> d_out
    gemm16<64, 64, false, true, 2><<<gemm_blk_64, B256, 0, stream>>>(
        h, W3, nullptr, b3, xl, NN);
    gemm16<64, 32, false, true, 0><<<gemm_blk_32, B256, 0, stream>>>(
        xl, W4, nullptr, b4, (float*)d_out, NN);
}